// LigerFusedLinearGRPOLoss_34024730919214
// MI455X (gfx1250) — compile-verified
//
#include <hip/hip_runtime.h>
#include <math.h>

// ---------------------------------------------------------------------------
// Fused Linear GRPO loss for MI455X (gfx1250, wave32, WMMA).
//
// Roofline: 2 GEMMs of 4096x2048x32000 fp32 = ~1.07 TFLOP vs ~590MB input
// traffic -> compute bound -> V_WMMA_F32_16X16X4_F32 (fp32 matrix pipe) so
// argmax over 32000 logits matches the fp32 reference. Logits
// ([4096,32000] = 512MB/model) are never materialized: streaming online-
// softmax row statistics instead. Branchless updates (no EXEC divergence),
// 4 independent WMMA accumulator chains for deeper VMEM pipelining, and
// global_prefetch of the next vocab strip.
// ---------------------------------------------------------------------------

typedef __attribute__((ext_vector_type(2))) float v2f;
typedef __attribute__((ext_vector_type(8))) float v8f;

#define B_    4
#define T_    1024
#define H_    2048
#define V_    32000
#define BT_   4096
#define BETA_ 0.1f
#define EPS_  1e-4f

#define WAVES  8            // waves per block (256 threads, wave32)
#define NTILE  (WAVES * 16) // 128 vocab columns per block iteration
#define VITERS (V_ / NTILE) // 250 (exact)

static __device__ __forceinline__ v8f wmma_f32_16x16x4(v2f a, v2f b, v8f c) {
  // 8-arg pattern: (neg_a, A, neg_b, B, c_mod, C, reuse_a, reuse_b)
  return __builtin_amdgcn_wmma_f32_16x16x4_f32(false, a, false, b, (short)0, c,
                                               false, false);
}

// ---------------------------------------------------------------------------
// GEMM + per-token row statistics.
// Grid: 256 blocks (one 16-token tile each), 256 threads (8 waves).
// A fragment (K=4 step): lane ln (0..15) of half h holds
//   A[row=tile*16+ln][k0+2h .. k0+2h+1]  -> contiguous float2 load.
// B fragment mirrors:  W[vbase+ln][k0+2h .. +1].
// C layout: VGPR r, half h -> token m = r + 8h, column n = ln.
// IS_REF=0: track max/argmax/sumexp/sumlogits.
// IS_REF=1: track max/sumexp and the logit at chosenIn[token].
// ---------------------------------------------------------------------------
template <int IS_REF>
__global__ __launch_bounds__(256) void grpo_gemm_rowstats(
    const float* __restrict__ X, const float* __restrict__ W,
    const int* __restrict__ chosenIn, int* __restrict__ chosenOut,
    float* __restrict__ clpOut, float* __restrict__ lseOut,
    float* __restrict__ slogOut, float* __restrict__ reflpOut) {
  const int tile = blockIdx.x;
  const int tid  = threadIdx.x;
  const int wave = tid >> 5;
  const int lane = tid & 31;
  const int h    = lane >> 4;   // lane half (K sub-pair / token-half select)
  const int ln   = lane & 15;

  const float* __restrict__ Arow = X + (size_t)(tile * 16 + ln) * H_ + 2 * h;

  float vmax[8], vsum[8], vext[8];
  int   varg[8], cidx[8];
#pragma unroll
  for (int r = 0; r < 8; ++r) {
    vmax[r] = -INFINITY; vsum[r] = 0.f; vext[r] = 0.f; varg[r] = 0; cidx[r] = -1;
  }
  if (IS_REF) {
#pragma unroll
    for (int r = 0; r < 8; ++r) cidx[r] = chosenIn[tile * 16 + r + 8 * h];
  }

  for (int j = 0; j < VITERS; ++j) {
    const int vbase = j * NTILE + wave * 16;
    const float* __restrict__ Brow = W + (size_t)(vbase + ln) * H_ + 2 * h;

    // Pull the next vocab strip's W rows toward L2 early
    // (lowers to global_prefetch_b8; A tile is already L2-resident).
    if (j + 1 < VITERS) {
      __builtin_prefetch(W + (size_t)(vbase + NTILE + ln) * H_, 0, 1);
    }

    // 4 independent accumulator chains -> 8 loads in flight per step,
    // no WMMA->WMMA RAW chain stalls.
    v8f c0 = {}, c1 = {}, c2 = {}, c3 = {};
#pragma unroll 2
    for (int k = 0; k < H_; k += 16) {
      v2f a0 = *(const v2f*)(Arow + k);
      v2f b0 = *(const v2f*)(Brow + k);
      v2f a1 = *(const v2f*)(Arow + k + 4);
      v2f b1 = *(const v2f*)(Brow + k + 4);
      v2f a2 = *(const v2f*)(Arow + k + 8);
      v2f b2 = *(const v2f*)(Brow + k + 8);
      v2f a3 = *(const v2f*)(Arow + k + 12);
      v2f b3 = *(const v2f*)(Brow + k + 12);
      c0 = wmma_f32_16x16x4(a0, b0, c0);
      c1 = wmma_f32_16x16x4(a1, b1, c1);
      c2 = wmma_f32_16x16x4(a2, b2, c2);
      c3 = wmma_f32_16x16x4(a3, b3, c3);
    }
    v8f c = (c0 + c1) + (c2 + c3);

    const int vcol = vbase + ln;  // vocab column owned by this lane
#pragma unroll
    for (int r = 0; r < 8; ++r) {
      const float val = c[r];  // logit(token = r + 8h, column = vcol)
      if (!IS_REF) {
        vext[r] += val;                                   // sum of logits
      } else {
        vext[r] += (vcol == cidx[r]) ? val : 0.f;         // chosen logit
      }
      // Branchless online softmax update (no EXEC divergence).
      const float nmax = fmaxf(vmax[r], val);
      vsum[r] = vsum[r] * __expf(vmax[r] - nmax) + __expf(val - nmax);
      varg[r] = (val > vmax[r]) ? vcol : varg[r];
      vmax[r] = nmax;
    }
  }

  // Reduce across the 16 lanes within each half (wave32 xor shuffles).
#pragma unroll
  for (int m = 1; m < 16; m <<= 1) {
#pragma unroll
    for (int r = 0; r < 8; ++r) {
      const float omax = __shfl_xor(vmax[r], m, 32);
      const float osum = __shfl_xor(vsum[r], m, 32);
      const float oext = __shfl_xor(vext[r], m, 32);
      const int   oarg = __shfl_xor(varg[r], m, 32);
      const float nmax = fmaxf(vmax[r], omax);
      vsum[r] = vsum[r] * __expf(vmax[r] - nmax) + osum * __expf(omax - nmax);
      const bool take = (omax > vmax[r]) | ((omax == vmax[r]) & (oarg < varg[r]));
      varg[r] = take ? oarg : varg[r];
      vmax[r] = nmax;
      vext[r] += oext;
    }
  }

  __shared__ float smax[WAVES][16];
  __shared__ float ssum[WAVES][16];
  __shared__ float sext[WAVES][16];
  __shared__ int   sarg[WAVES][16];
  if (ln == 0) {
#pragma unroll
    for (int r = 0; r < 8; ++r) {
      const int m = r + 8 * h;
      smax[wave][m] = vmax[r];
      ssum[wave][m] = vsum[r];
      sext[wave][m] = vext[r];
      sarg[wave][m] = varg[r];
    }
  }
  __syncthreads();

  if (tid < 16) {  // one thread per token: merge 8 wave-partials
    const int m = tid;
    float M = -INFINITY, S = 0.f, E = 0.f;
    int   A = 0x7fffffff;
    for (int w = 0; w < WAVES; ++w) {
      const float mx = smax[w][m], sm = ssum[w][m];
      const float nmax = fmaxf(M, mx);
      S = S * __expf(M - nmax) + sm * __expf(mx - nmax);
      if (mx > M || (mx == M && sarg[w][m] < A)) A = sarg[w][m];
      M = nmax;
      E += sext[w][m];
    }
    const int   g   = tile * 16 + m;
    const float lse = M + __logf(S);
    if (!IS_REF) {
      chosenOut[g] = A;
      clpOut[g]    = M - lse;  // chosen logprob = max - lse
      lseOut[g]    = lse;
      slogOut[g]   = E;        // sum of logits over V
    } else {
      reflpOut[g] = E - lse;   // ref logprob at chosen token
    }
  }
}

// ---------------------------------------------------------------------------
// Finalize: advantages, masked loss, metrics. One block, deterministic trees.
// ---------------------------------------------------------------------------
__global__ __launch_bounds__(256) void grpo_finalize(
    const int* __restrict__ mask, const float* __restrict__ rewards,
    const float* __restrict__ clp, const float* __restrict__ lse,
    const float* __restrict__ slog, const float* __restrict__ reflp,
    float* __restrict__ out) {
  const int tid = threadIdx.x;
  float clpSum = 0.f, clpSq = 0.f, lpAll = 0.f;
  float klB[4] = {0, 0, 0, 0}, mB[4] = {0, 0, 0, 0};

  for (int t = tid; t < BT_; t += 256) {
    const int   b  = t >> 10;  // T = 1024
    const float c  = clp[t];
    const float d  = reflp[t] - c;
    const float kl = __expf(d) - d - 1.f;
    clpSum += c;
    clpSq  += c * c;
    lpAll  += slog[t] - (float)V_ * lse[t];  // sum_v (logit - lse)
    if (mask[t] != 0) { klB[b] += kl; mB[b] += 1.f; }
  }

  __shared__ float red[256];
  __shared__ float tot[11];
  float vals[11] = {clpSum, clpSq, lpAll, klB[0], klB[1], klB[2], klB[3],
                    mB[0],  mB[1], mB[2], mB[3]};
  for (int q = 0; q < 11; ++q) {
    red[tid] = vals[q];
    __syncthreads();
    for (int s = 128; s > 0; s >>= 1) {
      if (tid < s) red[tid] += red[tid + s];
      __syncthreads();
    }
    if (tid == 0) tot[q] = red[0];
    __syncthreads();
  }

  if (tid == 0) {
    const float r0 = rewards[0], r1 = rewards[1], r2 = rewards[2], r3 = rewards[3];
    const float mean = 0.25f * (r0 + r1 + r2 + r3);
    const float var  = ((r0 - mean) * (r0 - mean) + (r1 - mean) * (r1 - mean) +
                        (r2 - mean) * (r2 - mean) + (r3 - mean) * (r3 - mean)) / 3.f;
    const float stdr = sqrtf(var);
    float adv[4];
    adv[0] = (r0 - mean) / (stdr + EPS_);
    adv[1] = (r1 - mean) / (stdr + EPS_);
    adv[2] = (r2 - mean) / (stdr + EPS_);
    adv[3] = (r3 - mean) / (stdr + EPS_);

    const float maskTot  = tot[7] + tot[8] + tot[9] + tot[10];
    const float klMasked = tot[3] + tot[4] + tot[5] + tot[6];
    float lossSum = BETA_ * klMasked;  // ratio==1 fwd -> policy term = -adv[b]
    for (int b = 0; b < 4; ++b) lossSum += -adv[b] * tot[7 + b];

    out[0] = lossSum / fmaxf(maskTot, 1.f);
    out[1] = tot[0] / (float)BT_;
    out[2] = sqrtf(fmaxf((tot[1] - tot[0] * tot[0] / (float)BT_) /
                             ((float)BT_ - 1.f), 0.f));
    out[3] = tot[2] / ((float)BT_ * (float)V_);
    out[4] = 0.25f * (tot[3] / tot[7] + tot[4] / tot[8] +
                      tot[5] / tot[9] + tot[6] / tot[10]);
  }
}

extern "C" void kernel_launch(void* const* d_in, const int* in_sizes, int n_in,
                              void* d_out, int out_size, void* d_ws,
                              size_t ws_size, hipStream_t stream) {
  const float* x     = (const float*)d_in[0];  // [B,T,H]
  const float* w     = (const float*)d_in[1];  // [V,H]
  const int*   amask = (const int*)d_in[2];    // [B,T]
  const float* rew   = (const float*)d_in[3];  // [B]
  const float* refx  = (const float*)d_in[4];  // [B,T,H]
  const float* refw  = (const float*)d_in[5];  // [V,H]
  float*       out   = (float*)d_out;          // [5]

  // Workspace layout (80 KB): chosen | clp | lse | slog | reflp, each 4096.
  char* ws = (char*)d_ws;
  int*   chosen = (int*)(ws);
  float* clp    = (float*)(ws + 16 * 1024);
  float* lse    = (float*)(ws + 32 * 1024);
  float* slog   = (float*)(ws + 48 * 1024);
  float* reflp  = (float*)(ws + 64 * 1024);

  dim3 grid(BT_ / 16), block(256);
  grpo_gemm_rowstats<0><<<grid, block, 0, stream>>>(
      x, w, nullptr, chosen, clp, lse, slog, nullptr);
  grpo_gemm_rowstats<1><<<grid, block, 0, stream>>>(
      refx, refw, chosen, nullptr, nullptr, nullptr, nullptr, reflp);
  grpo_finalize<<<dim3(1), block, 0, stream>>>(amask, rew, clp, lse, slog,
                                               reflp, out);
}